// MapAttention_46007689675366
// MI455X (gfx1250) — compile-verified
//
#include <hip/hip_runtime.h>

#define NBATCH   8
#define SEQ      1024
#define EMBED    128
#define HEADS    8
#define HEAD_DIM 16

typedef float v2f __attribute__((ext_vector_type(2)));
typedef float v8f __attribute__((ext_vector_type(8)));
typedef unsigned int v4u __attribute__((ext_vector_type(4)));
typedef unsigned int v8u __attribute__((ext_vector_type(8)));

// ---------------------------------------------------------------------------
// Kernel 1: fused per-head projections.
//   Qs[n,l,h*16+d] = (Wh[h]/sqrt(128)) * sum_e query[n,l,h*16+e] * Wq[d,e]
//   Ks[n,l,h*16+d] =                     sum_e keys [n,l,h*16+e] * Wk[d,e]
// ---------------------------------------------------------------------------
__global__ void proj_kernel(const float* __restrict__ query,
                            const float* __restrict__ keys,
                            const float* __restrict__ Wq,
                            const float* __restrict__ Wk,
                            const float* __restrict__ Wh,
                            float* __restrict__ Qs,
                            float* __restrict__ Ks) {
  int gid = blockIdx.x * blockDim.x + threadIdx.x;   // one thread per (row,col)
  int col = gid & (EMBED - 1);
  int row = gid >> 7;                                // 0 .. NBATCH*SEQ-1
  int h = col >> 4;
  int d = col & 15;
  const float inv_scale = 0.08838834764831845f;      // 1/sqrt(128)

  const float* qrow = query + (size_t)row * EMBED + h * HEAD_DIM;
  const float* krow = keys  + (size_t)row * EMBED + h * HEAD_DIM;
  const float* wq = Wq + d * HEAD_DIM;
  const float* wk = Wk + d * HEAD_DIM;

  float accq = 0.f, acck = 0.f;
#pragma unroll
  for (int e = 0; e < HEAD_DIM; ++e) {
    accq = fmaf(qrow[e], wq[e], accq);
    acck = fmaf(krow[e], wk[e], acck);
  }
  Qs[gid] = accq * (Wh[h] * inv_scale);
  Ks[gid] = acck;
}

// ---------------------------------------------------------------------------
// Kernel 2: batched GEMM  out[n] = Qs[n] (1024x128) @ Ks[n]^T (128x1024) + bh
// using V_WMMA_F32_16X16X4_F32.
//  - workgroup = 128x128 output tile (8 waves; wave w owns rows w*16..w*16+15)
//  - K-tile (128x128) DMA'd into LDS by the Tensor Data Mover, with hardware
//    row padding (+4 dwords per 128-dword row -> conflict-free ds_load_b64)
//  - A fragments preloaded from global concurrently with the TDM transfer
//  - two n-subtiles accumulated concurrently to break the WMMA RAW chain
// ---------------------------------------------------------------------------
__global__ void attn_gemm_kernel(const float* __restrict__ Qs,
                                 const float* __restrict__ Ks,
                                 const float* __restrict__ bh,
                                 float* __restrict__ out) {
  extern __shared__ float Bs[];            // 128 rows x (128+4) floats
  const int LDB = EMBED + 4;

  int bid  = blockIdx.x;
  int n    = bid >> 6;                     // batch
  int tile = bid & 63;
  int m0 = (tile >> 3) << 7;               // query-tile base
  int n0 = (tile & 7) << 7;                // key-tile base

  const float* Qb = Qs + (size_t)n * SEQ * EMBED;
  const float* Kb = Ks + (size_t)n * SEQ * EMBED;

  // ---- TDM stage: one tensor_load_to_lds per workgroup (wave 0 issues) ----
  if (threadIdx.x < 32) {
    unsigned long long ga = (unsigned long long)(const void*)(Kb + (size_t)n0 * EMBED);
    unsigned int lds0 = (unsigned int)(size_t)(void*)Bs;   // LDS byte offset

    v4u g0;
    g0[0] = 1u;                                            // count=1 (valid), user mode
    g0[1] = lds0;                                          // lds_addr
    g0[2] = (unsigned int)(ga & 0xffffffffu);              // global_addr[31:0]
    g0[3] = (unsigned int)((ga >> 32) & 0x01ffffffu)       // global_addr[56:32]
          | (2u << 30);                                    // type=2 ("image")

    v8u g1;
    g1[0] = (2u << 16)        // data_size = 4 bytes
          | (1u << 20)        // pad_enable
          | (6u << 22)        // pad_interval: every 128 dwords
          | (3u << 25);       // pad_amount: 4 dwords  -> LDS row pitch 132 dw
    g1[1] = (128u << 16);     // tensor_dim0 = 128 (lo16 in [31:16])
    g1[2] = (1024u << 16);    // tensor_dim0 hi = 0 ; tensor_dim1 lo = 1024
    g1[3] = (128u << 16);     // tensor_dim1 hi = 0 ; tile_dim0 = 128
    g1[4] = 128u;             // tile_dim1 = 128 ; tile_dim2 = 0
    g1[5] = 128u;             // tensor_dim0_stride = 128 (dim1_stride unused)
    g1[6] = 0u;
    g1[7] = 0u;

    asm volatile("tensor_load_to_lds %0, %1"
                 :: "s"(g0), "s"(g1)
                 : "memory");
  }

  int t     = threadIdx.x;
  int lane  = t & 31;
  int wave  = t >> 5;
  int rowl  = lane & 15;                   // M (A) / N (B) / N (C) index
  int half  = lane >> 4;                   // selects K pair {0,1} vs {2,3}
  int mbase = m0 + wave * 16;

  // Preload this wave's A fragments for all 32 K-chunks while the TDM DMA is
  // in flight (ISA 16x4 f32 layout: lane<16 -> K={0,1}, lane>=16 -> K={2,3},
  // row = lane&15).
  v2f a[32];
  const float* arow = Qb + (size_t)(mbase + rowl) * EMBED + 2 * half;
#pragma unroll
  for (int kk = 0; kk < 32; ++kk) {
    a[kk] = *(const v2f*)(arow + 4 * kk);
  }
  float bias = bh[0];

  // Tile DMA must be complete before any wave reads LDS.
  if (threadIdx.x < 32) {
    __builtin_amdgcn_s_wait_tensorcnt(0);
  }
  __syncthreads();

  for (int nt = 0; nt < 8; nt += 2) {
    v8f c0 = {0.f, 0.f, 0.f, 0.f, 0.f, 0.f, 0.f, 0.f};
    v8f c1 = {0.f, 0.f, 0.f, 0.f, 0.f, 0.f, 0.f, 0.f};
    const float* brow0 = Bs + (nt * 16 + rowl) * LDB + 2 * half;
    const float* brow1 = brow0 + 16 * LDB;
#pragma unroll
    for (int kk = 0; kk < 32; ++kk) {
      v2f b0 = *(const v2f*)(brow0 + 4 * kk);
      v2f b1 = *(const v2f*)(brow1 + 4 * kk);
      c0 = __builtin_amdgcn_wmma_f32_16x16x4_f32(
          false, a[kk], false, b0, (short)0, c0, false, false);
      c1 = __builtin_amdgcn_wmma_f32_16x16x4_f32(
          false, a[kk], false, b1, (short)0, c1, false, false);
    }
    // C layout: VGPR r -> row (r + 8*half), col = lane&15.
    float* orow0 = out + (size_t)n * SEQ * SEQ
                       + (size_t)(mbase + 8 * half) * SEQ
                       + (n0 + nt * 16 + rowl);
    float* orow1 = orow0 + 16;
#pragma unroll
    for (int r = 0; r < 8; ++r) {
      orow0[(size_t)r * SEQ] = c0[r] + bias;
      orow1[(size_t)r * SEQ] = c1[r] + bias;
    }
  }
}

// ---------------------------------------------------------------------------
extern "C" void kernel_launch(void* const* d_in, const int* in_sizes, int n_in,
                              void* d_out, int out_size, void* d_ws, size_t ws_size,
                              hipStream_t stream) {
  const float* query = (const float*)d_in[0];
  const float* keys  = (const float*)d_in[1];
  // d_in[2] = values  (dead in the reference)
  const float* Wq = (const float*)d_in[3];
  const float* Wk = (const float*)d_in[4];
  // d_in[5] = Wv      (dead in the reference)
  const float* Wh = (const float*)d_in[6];
  const float* bh = (const float*)d_in[7];
  float* out = (float*)d_out;

  float* Qs = (float*)d_ws;                               // 4 MB
  float* Ks = Qs + (size_t)NBATCH * SEQ * EMBED;          // 4 MB

  int total = NBATCH * SEQ * EMBED;                       // 1,048,576
  proj_kernel<<<total / 256, 256, 0, stream>>>(query, keys, Wq, Wk, Wh, Qs, Ks);

  size_t shmem = (size_t)128 * (EMBED + 4) * sizeof(float);  // 67,584 B
  attn_gemm_kernel<<<NBATCH * 64, 256, shmem, stream>>>(Qs, Ks, bh, out);
}